// Net_15212774162514
// MI455X (gfx1250) — compile-verified
//
#include <hip/hip_runtime.h>

#define KDIM 25
#define KD3  15625           // 25^3 kernel positions
#define BN_EPS 1e-5f

typedef unsigned int uint;
typedef __attribute__((ext_vector_type(4))) uint v4u;
typedef __attribute__((ext_vector_type(8))) int  v8i;
typedef __attribute__((ext_vector_type(4))) int  v4i;

#ifndef __has_builtin
#define __has_builtin(x) 0
#endif

#if __has_builtin(__builtin_amdgcn_tensor_load_to_lds)
#define HAVE_TDM 1
#else
#define HAVE_TDM 0
#endif

__device__ __forceinline__ int clampi(int v, int lo, int hi) {
  return v < lo ? lo : (v > hi ? hi : v);
}

// ---------------------------------------------------------------------------
// Kernel A: acc[n] = x[n,0]*root0 + x[n,1]*root1 + bias   (N = 100000)
// ---------------------------------------------------------------------------
__global__ void __launch_bounds__(256)
splineconv_init_acc(const float* __restrict__ x,
                    const float* __restrict__ root,
                    const float* __restrict__ bias,
                    float* __restrict__ acc, int n) {
  int i = blockIdx.x * blockDim.x + threadIdx.x;
  if (i < n) {
    float r0 = root[0], r1 = root[1], b = bias[0];
    acc[i] = fmaf(x[2 * i + 0], r0, fmaf(x[2 * i + 1], r1, b));
  }
}

// ---------------------------------------------------------------------------
// Kernel B: per-edge spline message + atomic scatter-add.
// Weight table (15625 x float2 = 125,000 B) is staged into LDS once per
// block via the Tensor Data Mover (wave 0 issues, waits TENSORcnt, barrier).
// Streamed read-once edge data (row/col/pseudo) uses non-temporal loads so
// the 192 MB L2 keeps x / acc / weight resident instead of the edge stream.
// ---------------------------------------------------------------------------
__global__ void __launch_bounds__(256)
splineconv_edge_kernel(const float* __restrict__ x,
                       const int* __restrict__ row,
                       const int* __restrict__ col,
                       const float* __restrict__ pseudo,
                       const float* __restrict__ weight,
                       float* __restrict__ acc,
                       int nedges) {
  extern __shared__ float2 sW[];   // KD3 float2 = 125,000 bytes dynamic LDS

#if HAVE_TDM
  if (threadIdx.x < 32u) {         // wave 0 only (wave32): one TDM op per block
    uint lds_lo = (uint)(unsigned long long)(&sW[0]);   // low 32b = LDS offset
    unsigned long long ga = (unsigned long long)weight;
    // ---- D# group 0 (128b): count=1 | lds_addr | global_addr | type=2 ----
    v4u g0 = {
      1u,                                               // count=1, user mode
      lds_lo,                                           // lds_addr [63:32]
      (uint)(ga & 0xFFFFFFFFull),                       // global_addr lo
      ((uint)((ga >> 32) & 0x01FFFFFFull)) | (2u << 30) // ga[56:32] | type=2
    };
    // ---- D# group 1 (256b): data_size=8B, 1-D tile of 15625 elements ----
    v8i g1 = {
      (int)(3u << 16),              // workgroup_mask=0, data_size=3 (8 bytes)
      (int)((uint)KD3 << 16),       // tensor_dim0[15:0]  (bits 63:48 of grp1)
      (int)(1u << 16),              // tensor_dim0 hi=0 | tensor_dim1=1
      (int)((uint)KD3 << 16),       // tile_dim0 = 15625
      1,                            // tile_dim1 = 1, tile_dim2 = 0 (unused)
      (int)KD3,                     // tensor_dim0_stride lo = 15625
      (int)((uint)KD3 << 16),       // dim0_stride hi=0 | dim1_stride lo16
      0                             // dim1_stride hi
    };
    v4i z4 = {0, 0, 0, 0};
#if __clang_major__ >= 23
    v8i z8 = {0, 0, 0, 0, 0, 0, 0, 0};
    __builtin_amdgcn_tensor_load_to_lds(g0, g1, z4, z4, z8, 0);
#else
    __builtin_amdgcn_tensor_load_to_lds(g0, g1, z4, z4, 0);
#endif
#if __has_builtin(__builtin_amdgcn_s_wait_tensorcnt)
    __builtin_amdgcn_s_wait_tensorcnt(0);
#else
    asm volatile("s_wait_tensorcnt 0x0" ::: "memory");
#endif
  }
#else
  // Fallback: cooperative staged copy
  for (int i = threadIdx.x; i < KD3; i += blockDim.x)
    sW[i] = ((const float2*)weight)[i];
#endif
  __syncthreads();

  const float2* __restrict__ x2 = (const float2*)x;
  const int stride = gridDim.x * blockDim.x;
  for (int e = blockIdx.x * blockDim.x + threadIdx.x; e < nedges; e += stride) {
    // pseudo in [0,1]^3 -> knot position in [0, K-1]; streamed once -> NT
    const float* pp = pseudo + 3 * e;
    float p0 = __builtin_nontemporal_load(pp + 0);
    float p1 = __builtin_nontemporal_load(pp + 1);
    float p2 = __builtin_nontemporal_load(pp + 2);

    float v0 = p0 * (float)(KDIM - 1);
    float v1 = p1 * (float)(KDIM - 1);
    float v2 = p2 * (float)(KDIM - 1);
    float f0 = floorf(v0), f1 = floorf(v1), f2 = floorf(v2);
    float t0 = v0 - f0, t1 = v1 - f1, t2 = v2 - f2;   // frac per dim

    int i0 = clampi((int)f0, 0, KDIM - 1), j0 = clampi((int)f0 + 1, 0, KDIM - 1);
    int i1 = clampi((int)f1, 0, KDIM - 1), j1 = clampi((int)f1 + 1, 0, KDIM - 1);
    int i2 = clampi((int)f2, 0, KDIM - 1), j2 = clampi((int)f2 + 1, 0, KDIM - 1);

    // pre-scale per-dim indices by strides {1, 25, 625}
    int a0 = i0,               a1 = j0;
    int b0 = KDIM * i1,        b1 = KDIM * j1;
    int c0 = KDIM * KDIM * i2, c1 = KDIM * KDIM * j2;
    float wx0 = 1.0f - t0, wx1 = t0;
    float wy0 = 1.0f - t1, wy1 = t1;
    float wz0 = 1.0f - t2, wz1 = t2;

    int src = __builtin_nontemporal_load(row + e);
    int dst = __builtin_nontemporal_load(col + e);
    float2 xs = x2[src];           // reused table: keep regular-temporal

    float msg = 0.0f;
#pragma unroll
    for (int s = 0; s < 8; ++s) {
      int   wi = ((s & 1) ? a1 : a0) + ((s & 2) ? b1 : b0) + ((s & 4) ? c1 : c0);
      float bw = ((s & 1) ? wx1 : wx0) * ((s & 2) ? wy1 : wy0) *
                 ((s & 4) ? wz1 : wz0);
      float2 wv = sW[wi];                       // ds_load_b64, LDS-resident
      msg = fmaf(bw, fmaf(xs.x, wv.x, xs.y * wv.y), msg);
    }
    atomicAdd(&acc[dst], msg);                  // global_atomic_add_f32 (no rtn)
  }
}

// ---------------------------------------------------------------------------
// Kernel C: out = gamma * (elu(acc) - mean) * rsqrt(var+eps) + beta
// ---------------------------------------------------------------------------
__global__ void __launch_bounds__(256)
splineconv_finalize(const float* __restrict__ acc,
                    const float* __restrict__ gamma,
                    const float* __restrict__ beta,
                    const float* __restrict__ rmean,
                    const float* __restrict__ rvar,
                    float* __restrict__ out, int n) {
  int i = blockIdx.x * blockDim.x + threadIdx.x;
  if (i < n) {
    float v = acc[i];
    v = (v > 0.0f) ? v : expm1f(v);             // ELU, alpha=1
    float scale = gamma[0] * rsqrtf(rvar[0] + BN_EPS);
    out[i] = fmaf(scale, v - rmean[0], beta[0]);
  }
}

// ---------------------------------------------------------------------------
extern "C" void kernel_launch(void* const* d_in, const int* in_sizes, int n_in,
                              void* d_out, int out_size, void* d_ws, size_t ws_size,
                              hipStream_t stream) {
  const float* x      = (const float*)d_in[0];
  const int*   eidx   = (const int*)d_in[1];   // int32 (JAX default x64 off)
  const float* pseudo = (const float*)d_in[2];
  const float* weight = (const float*)d_in[3]; // [15625, 2, 1] contiguous
  const float* root   = (const float*)d_in[4];
  const float* bias   = (const float*)d_in[5];
  const float* gamma  = (const float*)d_in[6];
  const float* beta   = (const float*)d_in[7];
  const float* rmean  = (const float*)d_in[8];
  const float* rvar   = (const float*)d_in[9];

  const int N = in_sizes[0] / 2;
  const int E = in_sizes[1] / 2;
  const int* row = eidx;        // edge_index[0]
  const int* col = eidx + E;    // edge_index[1]

  float* acc = (float*)d_ws;    // N floats of scratch accumulator
  float* out = (float*)d_out;

  const int tb = 256;           // 8 wave32 waves per block
  splineconv_init_acc<<<(N + tb - 1) / tb, tb, 0, stream>>>(x, root, bias, acc, N);

  const int blocks = 1024;      // ~25 edges/thread; 125 KB LDS -> 2 blocks/WGP
  const size_t shmem = (size_t)KD3 * sizeof(float2);   // 125,000 bytes
  splineconv_edge_kernel<<<blocks, tb, shmem, stream>>>(x, row, col, pseudo,
                                                        weight, acc, E);

  splineconv_finalize<<<(N + tb - 1) / tb, tb, 0, stream>>>(acc, gamma, beta,
                                                            rmean, rvar, out, N);
}